// TemporalGCN_55783035240724
// MI455X (gfx1250) — compile-verified
//
#include <hip/hip_runtime.h>

#define EPSF 1e-8f

typedef __attribute__((ext_vector_type(2))) float v2f;
typedef __attribute__((ext_vector_type(8))) float v8f;

// ---------------- utility kernels ----------------

__global__ void zero_f32_kernel(float* __restrict__ p, long long n) {
    long long i = (long long)blockIdx.x * blockDim.x + threadIdx.x;
    if (i < n) p[i] = 0.0f;
}

__global__ void init_minmax_kernel(unsigned int* mm) {
    mm[0] = 0x7F800000u;  // +inf  (min slot)
    mm[1] = 0u;           // 0.0f  (max slot) -- ts >= 0
}

__global__ void minmax_kernel(const float* __restrict__ ts, unsigned int* mm, int n) {
    int i = blockIdx.x * blockDim.x + threadIdx.x;
    if (i < n) {
        unsigned int b = __float_as_uint(ts[i]);  // ts in [0,1): bit order == float order
        atomicMin(&mm[0], b);
        atomicMax(&mm[1], b);
    }
}

__global__ void deg_kernel(const long long* __restrict__ dst, float* __restrict__ deg, int E) {
    int e = blockIdx.x * blockDim.x + threadIdx.x;
    if (e < E) atomicAdd(&deg[dst[e]], 1.0f);
}

__global__ void dinv_kernel(const float* __restrict__ deg, float* __restrict__ dinv, int n) {
    int i = blockIdx.x * blockDim.x + threadIdx.x;
    if (i < n) {
        float d = deg[i];
        dinv[i] = (d > 0.0f) ? rsqrtf(fmaxf(d, 1.0f)) : 0.0f;
    }
}

// copy x [N,128] into z1[:, 0:128] (row stride 512), float4 granularity
__global__ void copy_x_kernel(const float* __restrict__ x, float* __restrict__ z1, int N) {
    long long idx = (long long)blockIdx.x * blockDim.x + threadIdx.x;
    if (idx >= (long long)N * 32) return;
    int n = (int)(idx >> 5);
    int g = (int)(idx & 31);
    const float4 v = *(const float4*)(x + (long long)n * 128 + g * 4);
    *(float4*)(z1 + (long long)n * 512 + g * 4) = v;
}

// ---------------- CSR build: scan + counting sort ----------------
#define SCAN_B 256
#define SCAN_ELEMS 1024  // 4 per thread

__global__ __launch_bounds__(SCAN_B)
void scan_blocks_kernel(const float* __restrict__ deg, int* __restrict__ rowstart,
                        int* __restrict__ blocksums, int N) {
    __shared__ int sums[SCAN_B];
    const int t = threadIdx.x;
    const int base = blockIdx.x * SCAN_ELEMS + t * 4;
    int v[4];
    int s = 0;
#pragma unroll
    for (int j = 0; j < 4; ++j) {
        int i = base + j;
        v[j] = (i < N) ? (int)deg[i] : 0;
        s += v[j];
    }
    sums[t] = s;
    __syncthreads();
    // Hillis-Steele inclusive scan over per-thread sums
    for (int off = 1; off < SCAN_B; off <<= 1) {
        int x = sums[t];
        int y = (t >= off) ? sums[t - off] : 0;
        __syncthreads();
        sums[t] = x + y;
        __syncthreads();
    }
    int run = sums[t] - s;  // exclusive offset for this thread
#pragma unroll
    for (int j = 0; j < 4; ++j) {
        int i = base + j;
        if (i < N) rowstart[i] = run;
        run += v[j];
    }
    if (t == SCAN_B - 1) blocksums[blockIdx.x] = sums[t];
}

__global__ void scan_top_kernel(int* __restrict__ blocksums, int nb,
                                int* __restrict__ rowstart, int N, int E) {
    if (threadIdx.x == 0 && blockIdx.x == 0) {
        int run = 0;
        for (int b = 0; b < nb; ++b) {
            int t = blocksums[b];
            blocksums[b] = run;
            run += t;
        }
        rowstart[N] = E;
    }
}

__global__ void scan_add_kernel(int* __restrict__ rowstart, const int* __restrict__ blocksums, int N) {
    int i = blockIdx.x * blockDim.x + threadIdx.x;
    if (i < N) rowstart[i] += blocksums[i >> 10];  // SCAN_ELEMS == 1024
}

// counting-sort edges into CSR slots; pack {src, norm} as one float2 per edge
__global__ void fill_csr_kernel(const long long* __restrict__ src, const long long* __restrict__ dst,
                                const float* __restrict__ dinv, const int* __restrict__ rowstart,
                                int* __restrict__ cnt, float2* __restrict__ csr, int E) {
    int e = blockIdx.x * blockDim.x + threadIdx.x;
    if (e >= E) return;
    int s = (int)src[e];
    int d = (int)dst[e];
    int ofs = atomicAdd(&cnt[d], 1);
    int slot = rowstart[d] + ofs;
    float2 sw;
    sw.x = __int_as_float(s);
    sw.y = dinv[s] * dinv[d];
    csr[slot] = sw;
}

// ---------------- gather-form SpMM (no atomics) ----------------
// one wave per destination node; lane l owns features [4l..4l+3]
__global__ __launch_bounds__(256)
void spmm_gather_kernel(const float* __restrict__ hin, int in_stride,
                        float* __restrict__ hout, int out_stride,
                        const int* __restrict__ rowstart, const float2* __restrict__ csr, int N) {
    const int wave = threadIdx.x >> 5;
    const int lane = threadIdx.x & 31;
    const int node = blockIdx.x * 8 + wave;
    if (node >= N) return;
    const int jb = rowstart[node];
    const int je = rowstart[node + 1];
    float4 acc = make_float4(0.f, 0.f, 0.f, 0.f);
    const float* inb = hin + lane * 4;
    int j = jb;
    // 2 edges in flight per iteration
    for (; j + 2 <= je; j += 2) {
        float2 sw0 = csr[j];
        float2 sw1 = csr[j + 1];
        const float4 v0 = *(const float4*)(inb + (long long)__float_as_int(sw0.x) * in_stride);
        const float4 v1 = *(const float4*)(inb + (long long)__float_as_int(sw1.x) * in_stride);
        acc.x += v0.x * sw0.y; acc.y += v0.y * sw0.y;
        acc.z += v0.z * sw0.y; acc.w += v0.w * sw0.y;
        acc.x += v1.x * sw1.y; acc.y += v1.y * sw1.y;
        acc.z += v1.z * sw1.y; acc.w += v1.w * sw1.y;
    }
    if (j < je) {
        float2 sw = csr[j];
        const float4 v = *(const float4*)(inb + (long long)__float_as_int(sw.x) * in_stride);
        acc.x += v.x * sw.y; acc.y += v.y * sw.y;
        acc.z += v.z * sw.y; acc.w += v.w * sw.y;
    }
    *(float4*)(hout + (long long)node * out_stride + lane * 4) = acc;
}

// ---------------- dense GEMM via fp32 WMMA ----------------
// C[N,128] = relu?( A[N,Kdim] * W[Kdim,128] + bias )
// 256 threads = 8 waves; block covers 32 rows x 128 cols; each wave: two 16x16
// tiles (rows 0-15 / 16-31) sharing one B fragment -> halves W traffic per WMMA.
#define GEMM_THREADS 256
#define GEMM_ROWS 32
#define KC 256

__global__ __launch_bounds__(GEMM_THREADS)
void wmma_gemm_kernel(const float* __restrict__ A, int lda, int Kdim,
                      const float* __restrict__ W, const float* __restrict__ bias,
                      float* __restrict__ C, int ldc, int N, int do_relu) {
    __shared__ float As[GEMM_ROWS * (KC + 4)];  // pad 4 -> conflict-free frag reads
    const int Kpad = KC + 4;
    const int row0 = blockIdx.x * GEMM_ROWS;
    const int tid  = threadIdx.x;
    const int wave = tid >> 5;
    const int lane = tid & 31;
    const int m    = lane & 15;   // A row within tile / B,C col within tile
    const int half = lane >> 4;   // 0: K+{0,1}, 1: K+{2,3} per ISA 16x4 f32 layout
    const int col  = (wave << 4) + m;  // global output column (0..127)

    const float bv = bias[col];
    v8f acc0, acc1;
#pragma unroll
    for (int i = 0; i < 8; ++i) { acc0[i] = bv; acc1[i] = bv; }

    for (int k0 = 0; k0 < Kdim; k0 += KC) {
        const int kc = (Kdim - k0 < KC) ? (Kdim - k0) : KC;
        // stage 32 x kc A-chunk (clamp OOB rows; their stores are masked later)
        for (int idx = tid; idx < GEMM_ROWS * kc; idx += GEMM_THREADS) {
            int r = idx / kc;
            int c = idx - r * kc;
            int gr = row0 + r;
            if (gr >= N) gr = N - 1;
            As[r * Kpad + c] = A[(long long)gr * lda + k0 + c];
        }
        __syncthreads();
        // kc is always a multiple of 16 here -> unroll-4 remainder never runs
#pragma unroll 4
        for (int k = 0; k < kc; k += 4) {
            const int ko = k + 2 * half;
            const int kg = k0 + ko;
            v2f a0, a1, b;
            a0.x = As[m * Kpad + ko];
            a0.y = As[m * Kpad + ko + 1];
            a1.x = As[(16 + m) * Kpad + ko];
            a1.y = As[(16 + m) * Kpad + ko + 1];
            b.x  = W[(long long)kg * 128 + col];
            b.y  = W[(long long)(kg + 1) * 128 + col];
            acc0 = __builtin_amdgcn_wmma_f32_16x16x4_f32(false, a0, false, b, (short)0, acc0, false, false);
            acc1 = __builtin_amdgcn_wmma_f32_16x16x4_f32(false, a1, false, b, (short)0, acc1, false, false);
        }
        __syncthreads();
    }

    // C/D layout: VGPR i -> row (i + 8*half), col = lane&15
#pragma unroll
    for (int i = 0; i < 8; ++i) {
        int r = i + half * 8;
        int r0 = row0 + r;
        int r1 = row0 + 16 + r;
        float v0 = acc0[i];
        float v1 = acc1[i];
        if (do_relu) { v0 = fmaxf(v0, 0.0f); v1 = fmaxf(v1, 0.0f); }
        if (r0 < N) C[(long long)r0 * ldc + col] = v0;
        if (r1 < N) C[(long long)r1 * ldc + col] = v1;
    }
}

// ---------------- final head: out = [h2 | t*Wt+bt] @ Wf + bf ----------------
__global__ void final_kernel(const float* __restrict__ h2, const float* __restrict__ ts,
                             const unsigned int* __restrict__ mm,
                             const float* __restrict__ Wt, const float* __restrict__ bt,
                             const float* __restrict__ Wf, const float* __restrict__ bf,
                             float* __restrict__ out, int N) {
    int n = blockIdx.x * blockDim.x + threadIdx.x;
    if (n >= N) return;
    float tmin = __uint_as_float(mm[0]);
    float tmax = __uint_as_float(mm[1]);
    float t = (ts[n] - tmin) / (tmax - tmin + EPSF);
    float a0 = bf[0], a1 = bf[1];
    const float* hr = h2 + (long long)n * 128;
#pragma unroll 4
    for (int c = 0; c < 128; ++c) {
        float h = hr[c];
        a0 += h * Wf[c * 2 + 0];
        a1 += h * Wf[c * 2 + 1];
    }
#pragma unroll 4
    for (int c = 0; c < 128; ++c) {
        float te = t * Wt[c] + bt[c];
        a0 += te * Wf[(128 + c) * 2 + 0];
        a1 += te * Wf[(128 + c) * 2 + 1];
    }
    out[(long long)n * 2 + 0] = a0;
    out[(long long)n * 2 + 1] = a1;
}

// ---------------- launcher ----------------
extern "C" void kernel_launch(void* const* d_in, const int* in_sizes, int n_in,
                              void* d_out, int out_size, void* d_ws, size_t ws_size,
                              hipStream_t stream) {
    const float*     x   = (const float*)d_in[0];
    const long long* ei  = (const long long*)d_in[1];   // int64 [2,E] flat
    const float*     ts  = (const float*)d_in[2];
    const float*     W1  = (const float*)d_in[3];
    const float*     b1  = (const float*)d_in[4];
    const float*     W2  = (const float*)d_in[5];
    const float*     b2  = (const float*)d_in[6];
    const float*     Wt  = (const float*)d_in[7];
    const float*     bt  = (const float*)d_in[8];
    const float*     Wf  = (const float*)d_in[9];
    const float*     bf  = (const float*)d_in[10];

    const int N = in_sizes[2];          // ts has N elements
    const int E = in_sizes[1] / 2;
    const long long* src = ei;
    const long long* dst = ei + E;

    // workspace layout (4-byte words):
    // z1 (N*512) | z2 (N*384) | h2 (N*128) | csr (2E, 8B-aligned) |
    // deg (N) | cnt (N) | dinv (N) | rowstart (N+1) | blocksums (64) | mm (2)
    float* ws      = (float*)d_ws;
    float* z1      = ws;
    float* z2      = z1 + (long long)N * 512;
    float* h2      = z2 + (long long)N * 384;
    float2* csr    = (float2*)(h2 + (long long)N * 128);   // word offset N*1024 (even)
    float* deg     = (float*)(csr + E);
    int*   cnt     = (int*)(deg + N);
    float* dinv    = (float*)(cnt + N);
    int*   rowstart= (int*)(dinv + N);
    int*   bsums   = rowstart + (N + 1);
    unsigned int* mm = (unsigned int*)(bsums + 64);

    // only deg+cnt (adjacent) must be re-zeroed each call
    zero_f32_kernel<<<(2 * N + 255) / 256, 256, 0, stream>>>(deg, 2LL * N);
    init_minmax_kernel<<<1, 1, 0, stream>>>(mm);

    deg_kernel<<<(E + 255) / 256, 256, 0, stream>>>(dst, deg, E);
    dinv_kernel<<<(N + 255) / 256, 256, 0, stream>>>(deg, dinv, N);
    minmax_kernel<<<(N + 255) / 256, 256, 0, stream>>>(ts, mm, N);

    // CSR build
    const int nb = (N + SCAN_ELEMS - 1) / SCAN_ELEMS;
    scan_blocks_kernel<<<nb, SCAN_B, 0, stream>>>(deg, rowstart, bsums, N);
    scan_top_kernel<<<1, 1, 0, stream>>>(bsums, nb, rowstart, N, E);
    scan_add_kernel<<<(N + 255) / 256, 256, 0, stream>>>(rowstart, bsums, N);
    fill_csr_kernel<<<(E + 255) / 256, 256, 0, stream>>>(src, dst, dinv, rowstart, cnt, csr, E);

    copy_x_kernel<<<(unsigned)(((long long)N * 32 + 255) / 256), 256, 0, stream>>>(x, z1, N);

    const unsigned gg = (unsigned)((N + 7) / 8);  // 8 waves (nodes) per block
    // layer 1: K=3 hops, feats in z1 columns [0|128|256|384]
    spmm_gather_kernel<<<gg, 256, 0, stream>>>(z1 + 0,   512, z1 + 128, 512, rowstart, csr, N);
    spmm_gather_kernel<<<gg, 256, 0, stream>>>(z1 + 128, 512, z1 + 256, 512, rowstart, csr, N);
    spmm_gather_kernel<<<gg, 256, 0, stream>>>(z1 + 256, 512, z1 + 384, 512, rowstart, csr, N);
    // h = relu(z1 @ W1 + b1) -> written straight into z2[:, 0:128]
    wmma_gemm_kernel<<<(N + GEMM_ROWS - 1) / GEMM_ROWS, GEMM_THREADS, 0, stream>>>(
        z1, 512, 512, W1, b1, z2, 384, N, 1);
    // layer 2: K=2 hops in z2 columns [0|128|256]
    spmm_gather_kernel<<<gg, 256, 0, stream>>>(z2 + 0,   384, z2 + 128, 384, rowstart, csr, N);
    spmm_gather_kernel<<<gg, 256, 0, stream>>>(z2 + 128, 384, z2 + 256, 384, rowstart, csr, N);
    // h2 = relu(z2 @ W2 + b2)
    wmma_gemm_kernel<<<(N + GEMM_ROWS - 1) / GEMM_ROWS, GEMM_THREADS, 0, stream>>>(
        z2, 384, 384, W2, b2, h2, 128, N, 1);

    final_kernel<<<(N + 255) / 256, 256, 0, stream>>>(h2, ts, mm, Wt, bt, Wf, bf, (float*)d_out, N);
}